// ClassificationHead_79620103733804
// MI455X (gfx1250) — compile-verified
//
#include <hip/hip_runtime.h>
#include <hip/hip_bf16.h>
#include <math.h>

#define TASKS     64
#define N_WAY     5
#define N_SHOT    5
#define N_QUERY   75
#define N_SUPPORT 25
#define DDIM      16000
#define NVAR      125
#define C_REG_F   0.1f
#define SIGMA_F   0.1f
#define QP_ITERS  30

typedef float v2f __attribute__((ext_vector_type(2)));
typedef float v8f __attribute__((ext_vector_type(8)));

#define DC 128      // D-chunk staged in LDS per step (125 chunks exactly)
#define SW 132      // padded LDS row stride in floats (132 % 64 == 4 -> conflict-free)
#define SROWS 112   // rows 0..31: support (25 + pad), rows 32..111: query (75 + pad)

// ---------------------------------------------------------------------------
// Kernel 1: per-task Gram matrices via f32 WMMA.
//   Kws  [b][32][32] = support @ support^T   (valid 25x25)
//   Ksqws[b][32][80] = support @ query^T     (valid 25x75)
// One workgroup (8 waves) per task; 14 16x16 tiles over 8 waves:
//   wave w -> tile w (acc0), and tile w+8 for waves 0..5 (acc1).
// Tiles 0..3 : K    (mt = t>>1, nt = t&1),   B rows = support rows
// Tiles 4..13: Ksq  (u = t-4: mt = u&1, nt = u>>1), B rows = query rows (+32)
// ---------------------------------------------------------------------------
__global__ __launch_bounds__(256) void gram_kernel(
    const float* __restrict__ query, const float* __restrict__ support,
    float* __restrict__ Kws, float* __restrict__ Ksqws)
{
  __shared__ float sh[SROWS * SW];

  const int b    = blockIdx.x;
  const int tid  = threadIdx.x;
  const int wave = tid >> 5;
  const int lane = tid & 31;
  const float* Sg = support + (size_t)b * N_SUPPORT * DDIM;
  const float* Qg = query   + (size_t)b * N_QUERY   * DDIM;

  // zero pad rows once (support rows 25..31, query rows 107..111)
  for (int i = tid; i < 7 * SW; i += 256) sh[(25 + i / SW) * SW + i % SW] = 0.0f;
  for (int i = tid; i < 5 * SW; i += 256) sh[(107 + i / SW) * SW + i % SW] = 0.0f;

  // ---- per-wave tile descriptors (all wave-uniform scalars) ----
  // tile 0: t0 = wave
  int mrow0, brow0;
  if (wave < 4) { mrow0 = (wave >> 1) * 16; brow0 = (wave & 1) * 16; }
  else          { int u = wave - 4; mrow0 = (u & 1) * 16; brow0 = 32 + (u >> 1) * 16; }
  // tile 1: t1 = wave + 8 (valid for waves 0..5), always a Ksq tile (u = wave+4)
  const bool has1 = (wave < 6);
  const int u1    = wave + 4;
  const int mrow1 = (u1 & 1) * 16;
  const int brow1 = 32 + (u1 >> 1) * 16;

  const int kk = (lane >> 4) * 2;           // lanes 16..31 hold K = 2,3
  const int ml = lane & 15;                 // A: M index / B: N index

  const float* a0 = &sh[(mrow0 + ml) * SW]; // A rows always from support block
  const float* b0 = &sh[(brow0 + ml) * SW];
  const float* a1 = &sh[(mrow1 + ml) * SW];
  const float* b1 = &sh[(brow1 + ml) * SW];

  v8f acc0 = {};
  v8f acc1 = {};

  for (int d0 = 0; d0 < DDIM; d0 += DC) {
    __syncthreads();
    // stage support chunk (25 x 128 floats) as float4
    for (int i = tid; i < N_SUPPORT * (DC / 4); i += 256) {
      int r = i / (DC / 4), c = i % (DC / 4);
      float4 v = *(const float4*)(Sg + (size_t)r * DDIM + d0 + c * 4);
      *(float4*)(&sh[r * SW + c * 4]) = v;
    }
    // stage query chunk (75 x 128 floats) at row offset 32
    for (int i = tid; i < N_QUERY * (DC / 4); i += 256) {
      int r = i / (DC / 4), c = i % (DC / 4);
      float4 v = *(const float4*)(Qg + (size_t)r * DDIM + d0 + c * 4);
      *(float4*)(&sh[(32 + r) * SW + c * 4]) = v;
    }
    __syncthreads();

    #pragma unroll 8
    for (int k0 = 0; k0 < DC; k0 += 4) {
      v2f a, bb;
      a.x  = a0[k0 + kk];
      a.y  = a0[k0 + kk + 1];
      bb.x = b0[k0 + kk];
      bb.y = b0[k0 + kk + 1];
      acc0 = __builtin_amdgcn_wmma_f32_16x16x4_f32(
                 false, a, false, bb, (short)0, acc0, false, false);
    }
    if (has1) {
      #pragma unroll 8
      for (int k0 = 0; k0 < DC; k0 += 4) {
        v2f a, bb;
        a.x  = a1[k0 + kk];
        a.y  = a1[k0 + kk + 1];
        bb.x = b1[k0 + kk];
        bb.y = b1[k0 + kk + 1];
        acc1 = __builtin_amdgcn_wmma_f32_16x16x4_f32(
                   false, a, false, bb, (short)0, acc1, false, false);
      }
    }
  }

  // ---- write tiles: C/D layout -> VGPR v: lanes 0..15 M=v, lanes 16..31 M=v+8
  const int rl = (lane >> 4) * 8;
  const int cl = lane & 15;

  if (wave < 4) {            // tile 0 is a K tile
    const int ncol = brow0;  // == (wave&1)*16
    #pragma unroll
    for (int v = 0; v < 8; ++v)
      Kws[(size_t)b * 1024 + (size_t)(mrow0 + rl + v) * 32 + ncol + cl] = acc0[v];
  } else {                   // tile 0 is a Ksq tile
    const int ncol = brow0 - 32;
    #pragma unroll
    for (int v = 0; v < 8; ++v)
      Ksqws[(size_t)b * 2560 + (size_t)(mrow0 + rl + v) * 80 + ncol + cl] = acc0[v];
  }
  if (has1) {                // tile 1 is always a Ksq tile
    const int ncol = brow1 - 32;
    #pragma unroll
    for (int v = 0; v < 8; ++v)
      Ksqws[(size_t)b * 2560 + (size_t)(mrow1 + rl + v) * 80 + ncol + cl] = acc1[v];
  }
}

// ---------------------------------------------------------------------------
// Kernel 2: structure-exploiting primal-dual interior point QP + output.
// One block (5 waves = 160 threads) per task. KKT system solved via Schur
// complement over the block-diagonal H = blockdiag(K + diag(Dv_w)).
// ---------------------------------------------------------------------------
__global__ __launch_bounds__(160) void qp_kernel(
    const int* __restrict__ labels, const float* __restrict__ Kws,
    const float* __restrict__ Ksqws, float* __restrict__ out)
{
  __shared__ float Kmat[25][26];
  __shared__ float Lf[5][25][26];    // Cholesky factors of B_w
  __shared__ float Binv[5][25][26];  // B_w^{-1}
  __shared__ float Smat[25][26];     // Schur complement, factored in place
  __shared__ float ev[NVAR], hv[NVAR];
  __shared__ float z[NVAR], sv[NVAR], lamv[NVAR], nuv[25];
  __shared__ float Dv[NVAR], riv[NVAR], tmpv[NVAR], r1[NVAR], r2[25];
  __shared__ float yv[NVAR], dz[NVAR], dnu[25], dsvv[NVAR], dlv[NVAR];
  __shared__ float red[160];
  __shared__ float scal[2];          // [0]=mu, [1]=alpha

  const int b    = blockIdx.x;
  const int tid  = threadIdx.x;
  const int wave = tid >> 5;         // 0..4, each wave owns one K-block
  const int lane = tid & 31;

  for (int i = tid; i < 25 * 25; i += 160)
    Kmat[i / 25][i % 25] = Kws[(size_t)b * 1024 + (i / 25) * 32 + (i % 25)];
  for (int i = tid; i < NVAR; i += 160) {
    int w = i / 25, s = i % 25;
    float o = (labels[b * 25 + s] == w) ? 1.0f : 0.0f;
    ev[i] = -o; hv[i] = C_REG_F * o;
    z[i] = 0.0f; sv[i] = 1.0f; lamv[i] = 1.0f;
  }
  for (int i = tid; i < 25; i += 160) nuv[i] = 0.0f;
  __syncthreads();

  for (int iter = 0; iter < QP_ITERS; ++iter) {
    // ---- mu = dot(s, lam) / m ----
    float p = 0.0f;
    for (int i = tid; i < NVAR; i += 160) p += sv[i] * lamv[i];
    red[tid] = p; __syncthreads();
    if (tid == 0) { float a = 0.0f; for (int i = 0; i < 160; ++i) a += red[i]; scal[0] = a / 125.0f; }
    __syncthreads();
    const float mu = scal[0];

    // ---- residuals, Dv, tmp, rhs (Cmat = I simplifications) ----
    for (int i = tid; i < NVAR; i += 160) {
      int w = i / 25, s1 = i % 25;
      float gz = 0.0f;
      for (int s2 = 0; s2 < 25; ++s2) gz += Kmat[s1][s2] * z[w * 25 + s2];
      float rd  = gz + ev[i] + lamv[i] + nuv[s1];     // r_dual
      float ri  = z[i] + sv[i] - hv[i];               // r_ineq
      float dvi = lamv[i] / sv[i];
      float tm  = SIGMA_F * mu / sv[i] - lamv[i] + dvi * ri;
      Dv[i] = dvi; riv[i] = ri; tmpv[i] = tm;
      r1[i] = -(rd + tm);
    }
    for (int s = tid; s < 25; s += 160) {
      float a = 0.0f;
      for (int w = 0; w < 5; ++w) a += z[w * 25 + s];
      r2[s] = -a;                                     // -r_eq (b = 0)
    }
    __syncthreads();

    // ---- each wave: B_w = K + diag(Dv_w), Cholesky ----
    for (int i = lane; i < 25 * 25; i += 32)
      Lf[wave][i / 25][i % 25] =
          Kmat[i / 25][i % 25] + ((i / 25) == (i % 25) ? Dv[wave * 25 + i / 25] : 0.0f);
    __syncthreads();
    for (int k = 0; k < 25; ++k) {
      if (lane == 0) Lf[wave][k][k] = sqrtf(Lf[wave][k][k]);
      __syncthreads();
      float dkk = Lf[wave][k][k];
      if (lane > k && lane < 25) Lf[wave][lane][k] /= dkk;
      __syncthreads();
      if (lane > k && lane < 25) {
        float lik = Lf[wave][lane][k];
        for (int jj = k + 1; jj <= lane; ++jj)
          Lf[wave][lane][jj] -= lik * Lf[wave][jj][k];
      }
      __syncthreads();
    }

    // ---- B_w^{-1}: lane c solves B x = e_c (forward + back substitution) ----
    if (lane < 25) {
      float x[25];
      const int c = lane;
      for (int i = 0; i < 25; ++i) {
        float a = (i == c) ? 1.0f : 0.0f;
        for (int jj = 0; jj < i; ++jj) a -= Lf[wave][i][jj] * x[jj];
        x[i] = a / Lf[wave][i][i];
      }
      for (int i = 24; i >= 0; --i) {
        float a = x[i];
        for (int jj = i + 1; jj < 25; ++jj) a -= Lf[wave][jj][i] * x[jj];
        x[i] = a / Lf[wave][i][i];
      }
      for (int i = 0; i < 25; ++i) Binv[wave][i][c] = x[i];
    }
    __syncthreads();

    // ---- y_w = B_w^{-1} r1_w ----
    if (lane < 25) {
      float a = 0.0f;
      for (int c = 0; c < 25; ++c) a += Binv[wave][lane][c] * r1[wave * 25 + c];
      yv[wave * 25 + lane] = a;
    }
    __syncthreads();

    // ---- S = sum_w B_w^{-1}; rhs = sum_w y_w - r2 ----
    for (int i = tid; i < 25 * 25; i += 160) {
      float a = 0.0f;
      for (int w = 0; w < 5; ++w) a += Binv[w][i / 25][i % 25];
      Smat[i / 25][i % 25] = a;
    }
    for (int i = tid; i < 25; i += 160) {
      float a = 0.0f;
      for (int w = 0; w < 5; ++w) a += yv[w * 25 + i];
      dnu[i] = a - r2[i];
    }
    __syncthreads();

    // ---- Cholesky of S (wave 0 computes; barriers are block-uniform) ----
    for (int k = 0; k < 25; ++k) {
      if (tid == 0) Smat[k][k] = sqrtf(Smat[k][k]);
      __syncthreads();
      if (wave == 0 && lane > k && lane < 25) Smat[lane][k] /= Smat[k][k];
      __syncthreads();
      if (wave == 0 && lane > k && lane < 25) {
        float lik = Smat[lane][k];
        for (int jj = k + 1; jj <= lane; ++jj) Smat[lane][jj] -= lik * Smat[jj][k];
      }
      __syncthreads();
    }

    // ---- solve S dnu = rhs with wave32 shuffle substitution (wave 0) ----
    if (wave == 0) {
      float accv = (lane < 25) ? dnu[lane] : 0.0f;
      float xs[25];
      for (int jc = 0; jc < 25; ++jc) {
        float xj = __shfl(accv, jc, 32) / Smat[jc][jc];
        xs[jc] = xj;
        if (lane > jc && lane < 25) accv -= Smat[lane][jc] * xj;
      }
      float acc2 = (lane < 25) ? xs[lane] : 0.0f;
      float ymine = 0.0f;
      for (int jc = 24; jc >= 0; --jc) {
        float yj = __shfl(acc2, jc, 32) / Smat[jc][jc];
        if (lane == jc) ymine = yj;
        if (lane < jc) acc2 -= Smat[jc][lane] * yj;
      }
      if (lane < 25) dnu[lane] = ymine;
    }
    __syncthreads();

    // ---- dz_w = y_w - B_w^{-1} dnu ----
    if (lane < 25) {
      float corr = 0.0f;
      for (int c = 0; c < 25; ++c) corr += Binv[wave][lane][c] * dnu[c];
      dz[wave * 25 + lane] = yv[wave * 25 + lane] - corr;
    }
    __syncthreads();

    // ---- ds, dlam ----
    for (int i = tid; i < NVAR; i += 160) {
      dsvv[i] = -riv[i] - dz[i];
      dlv[i]  = tmpv[i] + Dv[i] * dz[i];
    }
    __syncthreads();

    // ---- fraction-to-boundary step length ----
    float mn = 1e9f;
    for (int i = tid; i < NVAR; i += 160) {
      float d1 = dsvv[i]; if (d1 < 0.0f) mn = fminf(mn, -sv[i]   / d1);
      float d2 = dlv[i];  if (d2 < 0.0f) mn = fminf(mn, -lamv[i] / d2);
    }
    red[tid] = mn; __syncthreads();
    if (tid == 0) {
      float a = 1e9f;
      for (int i = 0; i < 160; ++i) a = fminf(a, red[i]);
      scal[1] = fminf(1.0f, 0.99f * a);
    }
    __syncthreads();
    const float alpha = scal[1];
    for (int i = tid; i < NVAR; i += 160) {
      z[i]    += alpha * dz[i];
      sv[i]   += alpha * dsvv[i];
      lamv[i] += alpha * dlv[i];
    }
    for (int i = tid; i < 25; i += 160) nuv[i] += alpha * dnu[i];
    __syncthreads();
  }

  // ---- out[b][q][w] = sum_s alpha[w][s] * Ksq[s][q] ----
  for (int idx = tid; idx < N_QUERY * N_WAY; idx += 160) {
    int q = idx / N_WAY, w = idx % N_WAY;
    float a = 0.0f;
    for (int s = 0; s < 25; ++s)
      a += z[w * 25 + s] * Ksqws[(size_t)b * 2560 + (size_t)s * 80 + q];
    out[(size_t)b * N_QUERY * N_WAY + idx] = a;
  }
}

// ---------------------------------------------------------------------------
extern "C" void kernel_launch(void* const* d_in, const int* in_sizes, int n_in,
                              void* d_out, int out_size, void* d_ws, size_t ws_size,
                              hipStream_t stream) {
  const float* query   = (const float*)d_in[0];  // (64, 75, 16000) f32
  const float* support = (const float*)d_in[1];  // (64, 25, 16000) f32
  const int*   labels  = (const int*)d_in[2];    // (64, 25) i32

  float* Kws   = (float*)d_ws;                   // 64 * 32 * 32 f32
  float* Ksqws = Kws + (size_t)TASKS * 1024;     // 64 * 32 * 80 f32
  float* outp  = (float*)d_out;                  // (64, 75, 5) f32

  hipLaunchKernelGGL(gram_kernel, dim3(TASKS), dim3(256), 0, stream,
                     query, support, Kws, Ksqws);
  hipLaunchKernelGGL(qp_kernel, dim3(TASKS), dim3(160), 0, stream,
                     labels, Kws, Ksqws, outp);
}